// NCC_vxm_fast_81647328297394
// MI455X (gfx1250) — compile-verified
//
#include <hip/hip_runtime.h>

// ---------------------------------------------------------------------------
// Local NCC loss, 9x9x9 box window, volume (1,1,128,160,192), f32.
// 3-pass separable box filter; W and H passes run as banded-matrix matmuls
// on V_WMMA_F32_16X16X4_F32 (wave32, one 16x16 tile per wave).
// Channel index is grid-Y (SGPR-uniform); channel values are branchless.
// ---------------------------------------------------------------------------

typedef float v2f __attribute__((ext_vector_type(2)));
typedef float v8f __attribute__((ext_vector_type(8)));

constexpr int Dd = 128;
constexpr int Hh = 160;
constexpr int Ww = 192;
constexpr int ROWS = Dd * Hh;                           // 20480 rows/channel
constexpr long long CH_STRIDE = (long long)ROWS * Ww;   // 3,932,160 elems/ch
constexpr float WIN_INV = 1.0f / 729.0f;
constexpr float N_INV = 1.0f / 3932160.0f;

// Branchless channel value: ch0=I, ch1=J, ch2=I*I, ch3=J*J, ch4=I*J
__device__ __forceinline__ float chanval(float i, float j, int ch) {
  const float x = (ch == 1 || ch == 3) ? j : i;
  const float y = (ch <= 1) ? 1.0f : ((ch == 2) ? i : j);
  return x * y;
}

// ---------------------------------------------------------------------------
// Pass 1: box-sum along W (contiguous axis), fused with channel generation.
// Tile: M=16 rows, N=16 w_out, K in 6 chunks of 4 covering the 24-wide halo.
// A = data (16x4 f32), B = banded 0/1 matrix built in registers.
// Grid: x = 3840 blocks * 4 waves = 15360 tiles (1280 row-tiles x 12 w-tiles),
//       y = channel (uniform).
// ---------------------------------------------------------------------------
__global__ __launch_bounds__(128) void ncc_pass1_w(const float* __restrict__ I,
                                                   const float* __restrict__ J,
                                                   float* __restrict__ out) {
  const int lane  = threadIdx.x & 31;
  const int wave  = threadIdx.x >> 5;
  const int ch    = blockIdx.y;                // SGPR-uniform
  const int t     = blockIdx.x * 4 + wave;     // 15360 tiles per channel
  const int wt    = t % 12;
  const int rt    = t / 12;
  const int w0    = wt * 16;
  const int row0  = rt * 16;
  const int m     = lane & 15;                 // M index (A) / N index (B,D)
  const int khalf = lane >> 4;                 // K half selector

  const long long row = (long long)row0 + m;
  const float* Ir = I + row * Ww;
  const float* Jr = J + row * Ww;

  v8f acc = {0.f, 0.f, 0.f, 0.f, 0.f, 0.f, 0.f, 0.f};

#pragma unroll
  for (int c = 0; c < 6; ++c) {
    const int p0  = c * 4 + khalf * 2;         // K position within 24-halo
    const int wi0 = w0 - 4 + p0;
    // branchless clamped+masked loads (zero padding at row edges)
    const int wc0 = min(max(wi0, 0), Ww - 1);
    const int wc1 = min(max(wi0 + 1, 0), Ww - 1);
    const float m0 = (wi0 >= 0 && wi0 < Ww) ? 1.f : 0.f;
    const float m1 = (wi0 + 1 >= 0 && wi0 + 1 < Ww) ? 1.f : 0.f;
    const float a0 = chanval(Ir[wc0], Jr[wc0], ch) * m0;
    const float a1 = chanval(Ir[wc1], Jr[wc1], ch) * m1;
    v2f A = {a0, a1};
    // band: contributes iff m <= p <= m+8  (p = absolute halo position)
    v2f B = {(m <= p0 && p0 <= m + 8) ? 1.f : 0.f,
             (m <= p0 + 1 && p0 + 1 <= m + 8) ? 1.f : 0.f};
    acc = __builtin_amdgcn_wmma_f32_16x16x4_f32(false, A, false, B,
                                                (short)0, acc, false, false);
  }

  // D layout: lane -> N = m, VGPR v -> M = v + khalf*8
  float* Ob = out + (long long)ch * CH_STRIDE + (long long)row0 * Ww + w0;
  const int mr = khalf * 8;
#pragma unroll
  for (int v = 0; v < 8; ++v)
    Ob[(long long)(mr + v) * Ww + m] = acc[v];
}

// ---------------------------------------------------------------------------
// Pass 2: box-sum along H. Roles swapped so loads AND stores coalesce on W:
// Out[h_out][w] = Band^T[h_out][h_in] x In[h_in][w]
// A = band^T (built in regs), B = data (4x16, lanes = consecutive w).
// Grid: x = 3840 blocks * 4 waves = 15360 tiles (128 d x 10 h-tiles x 12
//       w-tiles), y = channel (uniform).
// ---------------------------------------------------------------------------
__global__ __launch_bounds__(128) void ncc_pass2_h(const float* __restrict__ in,
                                                   float* __restrict__ out) {
  const int lane  = threadIdx.x & 31;
  const int wave  = threadIdx.x >> 5;
  const int ch    = blockIdx.y;                // SGPR-uniform
  const int t     = blockIdx.x * 4 + wave;     // 15360 tiles per channel
  const int wt    = t % 12;
  const int rest  = t / 12;
  const int ht    = rest % 10;
  const int d     = rest / 10;
  const int w0    = wt * 16;
  const int h0    = ht * 16;
  const int m     = lane & 15;
  const int khalf = lane >> 4;

  const long long slab = ((long long)ch * Dd + d) * (long long)Hh * Ww;

  v8f acc = {0.f, 0.f, 0.f, 0.f, 0.f, 0.f, 0.f, 0.f};

#pragma unroll
  for (int c = 0; c < 6; ++c) {
    const int q0 = c * 4 + khalf * 2;
    // A = band^T: A[M=h_out=m][K] = (m <= q <= m+8)
    v2f A = {(m <= q0 && q0 <= m + 8) ? 1.f : 0.f,
             (m <= q0 + 1 && q0 + 1 <= m + 8) ? 1.f : 0.f};
    // B = data rows at h_in = h0-4+q (clamped+masked), lanes span w
    const int hi0 = h0 - 4 + q0;
    const int hc0 = min(max(hi0, 0), Hh - 1);
    const int hc1 = min(max(hi0 + 1, 0), Hh - 1);
    const float m0 = (hi0 >= 0 && hi0 < Hh) ? 1.f : 0.f;
    const float m1 = (hi0 + 1 >= 0 && hi0 + 1 < Hh) ? 1.f : 0.f;
    v2f B = {in[slab + (long long)hc0 * Ww + w0 + m] * m0,
             in[slab + (long long)hc1 * Ww + w0 + m] * m1};
    acc = __builtin_amdgcn_wmma_f32_16x16x4_f32(false, A, false, B,
                                                (short)0, acc, false, false);
  }

  // D: lane -> N = w offset m, VGPR v -> M = h_out offset v + khalf*8
  const int hr = h0 + khalf * 8;
#pragma unroll
  for (int v = 0; v < 8; ++v)
    out[slab + (long long)(hr + v) * Ww + w0 + m] = acc[v];
}

// ---------------------------------------------------------------------------
// Pass 3: box-sum along D (running 9-window, ring in registers), fused with
// the cc formula and wave reduction into a single float accumulator.
// One thread per (h,w) column; consecutive threads -> consecutive w.
// ---------------------------------------------------------------------------
__global__ __launch_bounds__(256) void ncc_pass3_d(const float* __restrict__ in,
                                                   float* __restrict__ acc) {
  const int idx = blockIdx.x * 256 + threadIdx.x;   // 0 .. 30719 (= H*W)
  const int w = idx % Ww;
  const int h = idx / Ww;
  const long long colBase = (long long)h * Ww + w;
  const long long dStride = (long long)Hh * Ww;

  float hist[5][9];
#pragma unroll
  for (int c = 0; c < 5; ++c)
#pragma unroll
    for (int k = 0; k < 9; ++k) hist[c][k] = 0.f;

  float s[5] = {0.f, 0.f, 0.f, 0.f, 0.f};
  float ccsum = 0.f;

  for (int t = 0; t < Dd + 4; ++t) {
    const int tc = (t < Dd) ? t : (Dd - 1);
    const float msk = (t < Dd) ? 1.f : 0.f;
    float nv[5];
#pragma unroll
    for (int c = 0; c < 5; ++c)
      nv[c] = in[(long long)c * CH_STRIDE + (long long)tc * dStride + colBase] * msk;

    __builtin_prefetch(in + (long long)min(t + 2, Dd - 1) * dStride + colBase, 0, 1);

#pragma unroll
    for (int c = 0; c < 5; ++c) {
      s[c] += nv[c] - hist[c][8];
#pragma unroll
      for (int k = 8; k > 0; --k) hist[c][k] = hist[c][k - 1];
      hist[c][0] = nv[c];
    }

    if (t >= 4) {   // s now covers window centered at d_out = t-4
      const float cross = s[4] - s[1] * s[0] * WIN_INV;
      const float ivar  = s[2] - s[0] * s[0] * WIN_INV;
      const float jvar  = s[3] - s[1] * s[1] * WIN_INV;
      ccsum += (cross * cross) / (ivar * jvar + 1e-5f);
    }
  }

  // wave32 reduction, then one atomic per wave
#pragma unroll
  for (int off = 16; off > 0; off >>= 1)
    ccsum += __shfl_down(ccsum, off, 32);
  if ((threadIdx.x & 31) == 0) atomicAdd(acc, ccsum);
}

__global__ void ncc_init(float* acc) { *acc = 0.f; }

__global__ void ncc_final(const float* __restrict__ acc, float* __restrict__ out) {
  out[0] = 1.0f - acc[0] * N_INV;
}

// ---------------------------------------------------------------------------
extern "C" void kernel_launch(void* const* d_in, const int* in_sizes, int n_in,
                              void* d_out, int out_size, void* d_ws, size_t ws_size,
                              hipStream_t stream) {
  (void)in_sizes; (void)n_in; (void)out_size; (void)ws_size;
  const float* I = (const float*)d_in[0];     // y_true
  const float* J = (const float*)d_in[1];     // y_pred
  float* out = (float*)d_out;

  float* wsA = (float*)d_ws;                  // 5ch W-summed   (78.6 MB)
  float* wsB = wsA + 5ll * CH_STRIDE;         // 5ch W+H-summed (78.6 MB)
  float* acc = wsB + 5ll * CH_STRIDE;         // scalar accumulator

  ncc_init<<<1, 1, 0, stream>>>(acc);
  // 15360 tiles/channel, 4 wave32 tiles per 128-thread block, grid.y = channel
  ncc_pass1_w<<<dim3(3840, 5), 128, 0, stream>>>(I, J, wsA);
  ncc_pass2_h<<<dim3(3840, 5), 128, 0, stream>>>(wsA, wsB);
  // 30720 columns, 256 threads/block
  ncc_pass3_d<<<120, 256, 0, stream>>>(wsB, acc);
  ncc_final<<<1, 1, 0, stream>>>(acc, out);
}